// GraphSAGE_29781303231030
// MI455X (gfx1250) — compile-verified
//
#include <hip/hip_runtime.h>

typedef __attribute__((ext_vector_type(2))) float v2f;
typedef __attribute__((ext_vector_type(8))) float v8f;

#define HDIM 128

// ---------------------------------------------------------------------------
// Degree accumulation: one thread per edge, atomicAdd 1.0 into deg[dst[e]].
// ---------------------------------------------------------------------------
__global__ void deg_kernel(const int* __restrict__ dst, float* __restrict__ deg, int E) {
  int e = blockIdx.x * blockDim.x + threadIdx.x;
  if (e < E) atomicAdd(&deg[dst[e]], 1.0f);
}

__global__ void invdeg_kernel(const float* __restrict__ deg, float* __restrict__ invd, int N) {
  int i = blockIdx.x * blockDim.x + threadIdx.x;
  if (i < N) {
    float d = deg[i];
    invd[i] = (d > 0.0f) ? (1.0f / fmaxf(d, 1.0f)) : 0.0f;
  }
}

// ---------------------------------------------------------------------------
// Edge scatter: one wave32 per edge. Lane loads float4 of the source feature
// row (global_load_b128) and atomically adds 4 floats into the destination
// row (global_atomic_add_f32 without return -> STOREcnt path).
// ---------------------------------------------------------------------------
__global__ void scatter_kernel(const float* __restrict__ feat,
                               const int* __restrict__ src,
                               const int* __restrict__ dst,
                               float* __restrict__ agg, int E) {
  int tid  = blockIdx.x * blockDim.x + threadIdx.x;
  int e    = tid >> 5;
  int lane = threadIdx.x & 31;
  if (e >= E) return;
  int s = src[e];
  int d = dst[e];
  const float4 v = *(const float4*)(feat + (size_t)s * HDIM + lane * 4);
  float* p = agg + (size_t)d * HDIM + lane * 4;
  atomicAdd(p + 0, v.x);
  atomicAdd(p + 1, v.y);
  atomicAdd(p + 2, v.z);
  atomicAdd(p + 3, v.w);
}

// ---------------------------------------------------------------------------
// Fused SAGE layer GEMM: out = relu?( (agg*invdeg) @ Wl + b + x @ Wr )
//   block  = 16 output rows, 8 waves
//   wave w = 16x16 output tile at columns [16w, 16w+16)
// V_WMMA_F32_16X16X4_F32, K swept 0..128 step 4, two WMMAs per step
// (neighbor-mean term and self term) chained through one f32 accumulator.
//
// 32-bit A 16x4 layout (ISA 7.12.2): lanes 0-15 -> M=lane, K=k0+{0,1};
// lanes 16-31 -> M=lane-16, K=k0+{2,3}. B (4x16) mirrors with lane=column.
// C/D: VGPR r holds M=r (lanes 0-15) / M=r+8 (lanes 16-31), N=lane%16.
// Epilogue bounds check is UNIFORM (full tile fast path) so EXEC stays
// all-ones and no divergent cmpx/branch chains are emitted.
// ---------------------------------------------------------------------------
template <bool RELU>
__global__ void sage_gemm_kernel(const float* __restrict__ x,
                                 const float* __restrict__ agg,
                                 const float* __restrict__ invd,
                                 const float* __restrict__ Wl,
                                 const float* __restrict__ Wr,
                                 const float* __restrict__ bias,
                                 float* __restrict__ out,
                                 int N) {
  const int lane = threadIdx.x & 31;
  const int wave = threadIdx.x >> 5;        // 0..7 : column tile
  const int row0 = blockIdx.x * 16;
  const int col0 = wave * 16;

  const int l16   = lane & 15;
  const int hi    = lane >> 4;              // 0 or 1
  const int koff  = hi * 2;                 // K sub-offset for this half-wave
  int mrow = row0 + l16;
  if (mrow >= N) mrow = N - 1;              // clamp (EXEC must stay all-ones)
  const float idg = invd[mrow];
  const int bcol = col0 + l16;

  const float* xrow = x   + (size_t)mrow * HDIM + koff;
  const float* arow = agg + (size_t)mrow * HDIM + koff;

  v8f acc = {};
  #pragma unroll 4
  for (int k0 = 0; k0 < HDIM; k0 += 4) {
    // A fragments
    v2f a_mean, a_self;
    a_mean.x = arow[k0 + 0] * idg;
    a_mean.y = arow[k0 + 1] * idg;
    a_self.x = xrow[k0 + 0];
    a_self.y = xrow[k0 + 1];
    // B fragments: W[k][col], row stride HDIM
    const float* wl = Wl + (size_t)(k0 + koff) * HDIM + bcol;
    const float* wr = Wr + (size_t)(k0 + koff) * HDIM + bcol;
    v2f b_l, b_r;
    b_l.x = wl[0];    b_l.y = wl[HDIM];
    b_r.x = wr[0];    b_r.y = wr[HDIM];

    acc = __builtin_amdgcn_wmma_f32_16x16x4_f32(false, a_mean, false, b_l,
                                                (short)0, acc, false, false);
    acc = __builtin_amdgcn_wmma_f32_16x16x4_f32(false, a_self, false, b_r,
                                                (short)0, acc, false, false);
  }

  // Epilogue: bias + optional ReLU, store per C/D VGPR layout.
  const float bv = bias[bcol];
  const int mbase = row0 + hi * 8;
  float* orow = out + (size_t)mbase * HDIM + bcol;

  if (row0 + 16 <= N) {                     // uniform: full tile (always for N%16==0)
    #pragma unroll
    for (int r = 0; r < 8; ++r) {
      float v = acc[r] + bv;
      if (RELU) v = fmaxf(v, 0.0f);
      orow[(size_t)r * HDIM] = v;
    }
  } else {                                  // ragged last tile (not hit when N%16==0)
    #pragma unroll
    for (int r = 0; r < 8; ++r) {
      if (mbase + r < N) {
        float v = acc[r] + bv;
        if (RELU) v = fmaxf(v, 0.0f);
        orow[(size_t)r * HDIM] = v;
      }
    }
  }
}

// ---------------------------------------------------------------------------
// Launcher
// d_in order: x, edge_index, Wl1, b1, Wr1, Wl2, b2, Wr2, Wl3, b3, Wr3
// ---------------------------------------------------------------------------
extern "C" void kernel_launch(void* const* d_in, const int* in_sizes, int n_in,
                              void* d_out, int out_size, void* d_ws, size_t ws_size,
                              hipStream_t stream) {
  const float* x   = (const float*)d_in[0];
  const int*   ei  = (const int*)d_in[1];
  const float* Wl1 = (const float*)d_in[2];
  const float* b1  = (const float*)d_in[3];
  const float* Wr1 = (const float*)d_in[4];
  const float* Wl2 = (const float*)d_in[5];
  const float* b2  = (const float*)d_in[6];
  const float* Wr2 = (const float*)d_in[7];
  const float* Wl3 = (const float*)d_in[8];
  const float* b3  = (const float*)d_in[9];
  const float* Wr3 = (const float*)d_in[10];

  const int N = in_sizes[0] / HDIM;
  const int E = in_sizes[1] / 2;
  const int* src = ei;
  const int* dst = ei + E;

  // Workspace partition (256B-aligned slices)
  char* ws = (char*)d_ws;
  size_t featBytes = ((size_t)N * HDIM * sizeof(float) + 255) & ~(size_t)255;
  size_t vecBytes  = ((size_t)N * sizeof(float) + 255) & ~(size_t)255;
  float* agg  = (float*)(ws);
  float* h1   = (float*)(ws + featBytes);
  float* h2   = (float*)(ws + 2 * featBytes);
  float* deg  = (float*)(ws + 3 * featBytes);
  float* invd = (float*)(ws + 3 * featBytes + vecBytes);

  const int TPB = 256;
  const int edgeBlocks    = (E + TPB - 1) / TPB;
  const int nodeBlocks    = (N + TPB - 1) / TPB;
  const int scatterBlocks = (int)(((size_t)E * 32 + TPB - 1) / TPB);
  const int gemmBlocks    = (N + 15) / 16;

  // Degrees (recomputed every call: deterministic, graph-capture safe)
  hipMemsetAsync(deg, 0, (size_t)N * sizeof(float), stream);
  deg_kernel<<<edgeBlocks, TPB, 0, stream>>>(dst, deg, E);
  invdeg_kernel<<<nodeBlocks, TPB, 0, stream>>>(deg, invd, N);

  // Layer 1: x -> h1 (ReLU)
  hipMemsetAsync(agg, 0, (size_t)N * HDIM * sizeof(float), stream);
  scatter_kernel<<<scatterBlocks, TPB, 0, stream>>>(x, src, dst, agg, E);
  sage_gemm_kernel<true><<<gemmBlocks, TPB, 0, stream>>>(x, agg, invd, Wl1, Wr1, b1, h1, N);

  // Layer 2: h1 -> h2 (ReLU)
  hipMemsetAsync(agg, 0, (size_t)N * HDIM * sizeof(float), stream);
  scatter_kernel<<<scatterBlocks, TPB, 0, stream>>>(h1, src, dst, agg, E);
  sage_gemm_kernel<true><<<gemmBlocks, TPB, 0, stream>>>(h1, agg, invd, Wl2, Wr2, b2, h2, N);

  // Layer 3: h2 -> out (no ReLU)
  hipMemsetAsync(agg, 0, (size_t)N * HDIM * sizeof(float), stream);
  scatter_kernel<<<scatterBlocks, TPB, 0, stream>>>(h2, src, dst, agg, E);
  sage_gemm_kernel<false><<<gemmBlocks, TPB, 0, stream>>>(h2, agg, invd, Wl3, Wr3, b3,
                                                          (float*)d_out, N);
}